// NVectorAction_41437844472221
// MI455X (gfx1250) — compile-verified
//
#include <hip/hip_runtime.h>

// XY/Heisenberg action:  out[s] = 2*V - sum_{v} S_v · (S_up(v) + S_left(v))
// with S = (cos th, sin th cos ph, sin th sin ph), periodic 256x256 lattice.
// Memory-bound streaming reduction: 134 MB in, 1 KB out.

namespace {
constexpr int   kL            = 256;
constexpr int   kVol          = kL * kL;
constexpr float kBeta         = 1.0f;
constexpr float kActionShift  = 2.0f * kBeta * (float)kVol;

constexpr int kWaves       = 8;             // waves per block (one block per state)
constexpr int kThreads     = kWaves * 32;   // 256
constexpr int kRowsPerWave = kL / kWaves;   // 32 lattice rows per wave
constexpr int kColsPerLane = kL / 32;       // 8 columns per lane (one row per wave)
} // namespace

// ---- CDNA5 async global->LDS path (probe via __has_builtin; fallback = direct loads) ----
#if defined(__HIP_DEVICE_COMPILE__) && defined(__gfx1250__) &&                 \
    __has_builtin(__builtin_amdgcn_global_load_async_to_lds_b128) &&           \
    __has_builtin(__builtin_amdgcn_s_wait_asynccnt)
#define USE_ASYNC_LDS 1
#else
#define USE_ASYNC_LDS 0
#endif

#if USE_ASYNC_LDS
// Exact parameter types per clang-22 diagnostic:
//   arg0: v4i32 in AS1 (printed as "int __vector(4) __device__ *")
//   arg1: v4i32 in AS3
typedef int v4i __attribute__((vector_size(16)));
typedef __attribute__((address_space(1))) v4i* gvec_t;
typedef __attribute__((address_space(3))) v4i* lvec_t;

// Copy one 1KB lattice row (theta) + one 1KB row (phi) into LDS, asynchronously.
// 32 lanes x b128 = 512B per instruction -> 4 instructions total, all on ASYNCcnt.
__device__ __forceinline__ void issue_row_async(const float* gt, const float* gp,
                                                float* lt, float* lp, int lane) {
  const __attribute__((address_space(1))) float* g0 =
      (const __attribute__((address_space(1))) float*)gt;
  const __attribute__((address_space(1))) float* g1 =
      (const __attribute__((address_space(1))) float*)gp;
  __attribute__((address_space(3))) float* l0 =
      (__attribute__((address_space(3))) float*)lt;
  __attribute__((address_space(3))) float* l1 =
      (__attribute__((address_space(3))) float*)lp;
  const int off = lane * 4;  // 16B per lane
  __builtin_amdgcn_global_load_async_to_lds_b128((gvec_t)(g0 + off),       (lvec_t)(l0 + off),       0, 0);
  __builtin_amdgcn_global_load_async_to_lds_b128((gvec_t)(g0 + off + 128), (lvec_t)(l0 + off + 128), 0, 0);
  __builtin_amdgcn_global_load_async_to_lds_b128((gvec_t)(g1 + off),       (lvec_t)(l1 + off),       0, 0);
  __builtin_amdgcn_global_load_async_to_lds_b128((gvec_t)(g1 + off + 128), (lvec_t)(l1 + off + 128), 0, 0);
}
#endif

__device__ __forceinline__ void spin3(float th, float ph, float& a, float& b, float& c) {
  a = __cosf(th);
  const float st = __sinf(th);
  b = st * __cosf(ph);
  c = st * __sinf(ph);
}

__global__ __launch_bounds__(kThreads)
void nvec_action_kernel(const float* __restrict__ state, float* __restrict__ out) {
  const int s    = blockIdx.x;
  const int tid  = threadIdx.x;
  const int wave = tid >> 5;
  const int lane = tid & 31;
  const int col0 = lane * kColsPerLane;

  const float* theta = state + (size_t)s * (size_t)(2 * kVol);
  const float* phi   = theta + kVol;
  const int r0 = wave * kRowsPerWave;

  // Prologue: S of the row above this strip (periodic wrap); kept in registers.
  float pa[kColsPerLane], pb[kColsPerLane], pc[kColsPerLane];
  {
    const int pr = (r0 + kL - 1) & (kL - 1);
    const float* trow = theta + pr * kL + col0;
    const float* prow = phi   + pr * kL + col0;
    float th[kColsPerLane], ph[kColsPerLane];
    *(float4*)&th[0] = *(const float4*)(trow);
    *(float4*)&th[4] = *(const float4*)(trow + 4);
    *(float4*)&ph[0] = *(const float4*)(prow);
    *(float4*)&ph[4] = *(const float4*)(prow + 4);
#pragma unroll
    for (int j = 0; j < kColsPerLane; ++j) spin3(th[j], ph[j], pa[j], pb[j], pc[j]);
  }

  float acc = 0.0f;

#if USE_ASYNC_LDS
  // Per-wave double buffers: 8 waves * 2 * (1KB theta + 1KB phi) = 32 KB LDS.
  __shared__ __align__(16) float bufT[kWaves][2][kL];
  __shared__ __align__(16) float bufP[kWaves][2][kL];
  issue_row_async(theta + r0 * kL, phi + r0 * kL,
                  &bufT[wave][0][0], &bufP[wave][0][0], lane);
#endif

  for (int i = 0; i < kRowsPerWave; ++i) {
    float th[kColsPerLane], ph[kColsPerLane];
#if USE_ASYNC_LDS
    __builtin_amdgcn_s_wait_asynccnt(0);                 // row i resident in LDS
    if (i + 1 < kRowsPerWave) {                          // overlap: fetch row i+1
      const int rn = r0 + i + 1;
      issue_row_async(theta + rn * kL, phi + rn * kL,
                      &bufT[wave][(i + 1) & 1][0], &bufP[wave][(i + 1) & 1][0], lane);
    }
    const float* lt = &bufT[wave][i & 1][col0];
    const float* lp = &bufP[wave][i & 1][col0];
    *(float4*)&th[0] = *(const float4*)(lt);
    *(float4*)&th[4] = *(const float4*)(lt + 4);
    *(float4*)&ph[0] = *(const float4*)(lp);
    *(float4*)&ph[4] = *(const float4*)(lp + 4);
#else
    const int rr = r0 + i;
    const float* trow = theta + rr * kL + col0;
    const float* prow = phi   + rr * kL + col0;
    if (i + 1 < kRowsPerWave) {                          // gfx1250 global_prefetch_b8
      __builtin_prefetch(trow + kL, 0, 1);
      __builtin_prefetch(prow + kL, 0, 1);
    }
    *(float4*)&th[0] = *(const float4*)(trow);
    *(float4*)&th[4] = *(const float4*)(trow + 4);
    *(float4*)&ph[0] = *(const float4*)(prow);
    *(float4*)&ph[4] = *(const float4*)(prow + 4);
#endif

    float a[kColsPerLane], b[kColsPerLane], c[kColsPerLane];
#pragma unroll
    for (int j = 0; j < kColsPerLane; ++j) spin3(th[j], ph[j], a[j], b[j], c[j]);

    // Vertical coupling: S[i,j] · S[i-1,j] (same lane, same column, registers).
#pragma unroll
    for (int j = 0; j < kColsPerLane; ++j) {
      acc = fmaf(a[j], pa[j], acc);
      acc = fmaf(b[j], pb[j], acc);
      acc = fmaf(c[j], pc[j], acc);
    }

    // Horizontal coupling: S[i,j] · S[i,j-1]. Column col0-1 lives in lane-1
    // (wave32 rotate handles the periodic j wrap: lane 0 <- lane 31's col 255).
    const int src = (lane + 31) & 31;
    const float aL = __shfl(a[kColsPerLane - 1], src, 32);
    const float bL = __shfl(b[kColsPerLane - 1], src, 32);
    const float cL = __shfl(c[kColsPerLane - 1], src, 32);
    acc = fmaf(a[0], aL, acc);
    acc = fmaf(b[0], bL, acc);
    acc = fmaf(c[0], cL, acc);
#pragma unroll
    for (int j = 1; j < kColsPerLane; ++j) {
      acc = fmaf(a[j], a[j - 1], acc);
      acc = fmaf(b[j], b[j - 1], acc);
      acc = fmaf(c[j], c[j - 1], acc);
    }

#pragma unroll
    for (int j = 0; j < kColsPerLane; ++j) { pa[j] = a[j]; pb[j] = b[j]; pc[j] = c[j]; }
  }

  // Deterministic reduction: wave32 shfl-xor tree, then per-block sum in LDS.
#pragma unroll
  for (int o = 16; o > 0; o >>= 1) acc += __shfl_xor(acc, o, 32);

  __shared__ float wsum[kWaves];
  if (lane == 0) wsum[wave] = acc;
  __syncthreads();
  if (tid == 0) {
    float tot = 0.0f;
#pragma unroll
    for (int w = 0; w < kWaves; ++w) tot += wsum[w];
    out[s] = kActionShift - kBeta * tot;  // -beta * sum + 2*beta*V
  }
}

extern "C" void kernel_launch(void* const* d_in, const int* in_sizes, int n_in,
                              void* d_out, int out_size, void* d_ws, size_t ws_size,
                              hipStream_t stream) {
  (void)n_in; (void)d_ws; (void)ws_size; (void)out_size;
  const float* state = (const float*)d_in[0];
  // d_in[1] (shift) unused: roll-by-1 neighbor indices are derived analytically.
  float* out = (float*)d_out;
  const int n_states = in_sizes[0] / (2 * kVol);
  nvec_action_kernel<<<n_states, kThreads, 0, stream>>>(state, out);
}